// Damped_electrostatics_ShiftedForce_48498770706888
// MI455X (gfx1250) — compile-verified
//
#include <hip/hip_runtime.h>
#include <math.h>

// Damped shifted-force electrostatics over 3.2M edges (memory-bound).
// Streams (vec/dist/idx) are staged into LDS with CDNA5 async global->LDS
// copies (ASYNCcnt) in a double-buffered pipeline; atom tables (5.2 MB,
// L2-resident) are gathered directly. No matrix-shaped math -> no WMMA.
// All stream offsets fit in u32 (max 38.4 MB), so stage math is 32-bit.

#define KEHALF 7.199822675975274f
#define TILE 256

// One async dword copy from global sbase plus byte_off to LDS at lds_dst (GVS addressing)
__device__ __forceinline__ void async_ld_dword(const void* lds_dst,
                                               const void* sbase,
                                               unsigned byte_off) {
    const unsigned lds = (unsigned)(unsigned long long)lds_dst;  // low 32 bits of the pointer are the LDS offset
    asm volatile("global_load_async_to_lds_b32 %0, %1, %2"
                 :: "v"(lds), "v"(byte_off), "s"(sbase)
                 : "memory");
}

// Stage one 256-edge tile (exactly 6 async instructions per lane, full EXEC:
// out-of-range lanes are clamped to the last valid dword so per-wave ASYNCcnt
// bookkeeping stays deterministic; clamped values are never consumed).
__device__ __forceinline__ void stage_tile(
    float* sv, float* sd, int* siu, int* siv,
    const float* vec, const float* dist, const int* iu, const int* iv,
    unsigned e0, unsigned n_edges, unsigned tid)
{
    const unsigned vmaxb = 3u * n_edges * 4u - 4u;   // last valid byte off in vec
    const unsigned emaxb = n_edges * 4u - 4u;        // last valid byte off in [E] arrays

    unsigned g0 = (3u * e0 + tid) * 4u;              // dword byte offsets into vec
    unsigned g1 = g0 + TILE * 4u;
    unsigned g2 = g1 + TILE * 4u;
    g0 = (g0 > vmaxb) ? vmaxb : g0;
    g1 = (g1 > vmaxb) ? vmaxb : g1;
    g2 = (g2 > vmaxb) ? vmaxb : g2;
    unsigned ge = (e0 + tid) * 4u;
    ge = (ge > emaxb) ? emaxb : ge;

    async_ld_dword(sv + tid,            vec,  g0);
    async_ld_dword(sv + tid + TILE,     vec,  g1);
    async_ld_dword(sv + tid + 2 * TILE, vec,  g2);
    async_ld_dword(sd + tid,            dist, ge);
    async_ld_dword(siu + tid,           iu,   ge);
    async_ld_dword(siv + tid,           iv,   ge);
}

__global__ __launch_bounds__(256) void edge_elec_kernel(
    const float* __restrict__ charges,     // [N]
    const float* __restrict__ dipoles,     // [N,3]
    const float* __restrict__ quads,       // [N,3,3]
    const float* __restrict__ vec,         // [E,3]  streamed via async->LDS
    const float* __restrict__ dist,        // [E]    streamed via async->LDS
    const int*   __restrict__ idx_u,       // [E]    streamed via async->LDS
    const int*   __restrict__ idx_v,       // [E]    streamed via async->LDS
    float*       __restrict__ out,         // [E]    non-temporal store
    int n_edges)
{
    __shared__ float sv[2][3 * TILE];
    __shared__ float sd[2][TILE];
    __shared__ int   siu[2][TILE];
    __shared__ int   siv[2][TILE];

    const unsigned tid    = threadIdx.x;
    const int      ntiles = (n_edges + TILE - 1) / TILE;
    int tb = (int)blockIdx.x;
    if (tb >= ntiles) return;                       // uniform per block

    stage_tile(sv[0], sd[0], siu[0], siv[0],
               vec, dist, idx_u, idx_v, (unsigned)tb * TILE, (unsigned)n_edges, tid);

    int buf = 0;
    for (; tb < ntiles; tb += (int)gridDim.x) {
        const int  nt        = tb + (int)gridDim.x;
        const bool have_next = nt < ntiles;
        if (have_next) {
            stage_tile(sv[buf ^ 1], sd[buf ^ 1], siu[buf ^ 1], siv[buf ^ 1],
                       vec, dist, idx_u, idx_v,
                       (unsigned)nt * TILE, (unsigned)n_edges, tid);
            // 6 newer ops may remain in flight; the 6 for tile tb are done.
            asm volatile("s_wait_asynccnt 0x6" ::: "memory");
        } else {
            asm volatile("s_wait_asynccnt 0x0" ::: "memory");
        }
        __syncthreads();                            // tile tb visible to all waves

        const int e = tb * TILE + (int)tid;
        if (e < n_edges) {
            const float d  = sd[buf][tid];
            const float vx = sv[buf][3 * tid + 0];
            const float vy = sv[buf][3 * tid + 1];
            const float vz = sv[buf][3 * tid + 2];
            const int   u  = siu[buf][tid];
            const int   v  = siv[buf][tid];

            // gathers from small L2-resident tables
            const float qu = charges[u];
            const float qv = charges[v];
            const float dux = dipoles[3 * u + 0];
            const float duy = dipoles[3 * u + 1];
            const float duz = dipoles[3 * u + 2];
            const float dvx = dipoles[3 * v + 0];
            const float dvy = dipoles[3 * v + 1];
            const float dvz = dipoles[3 * v + 2];
            const float* Q = quads + 9 * v;
            const float q00 = Q[0], q01 = Q[1], q02 = Q[2];
            const float q10 = Q[3], q11 = Q[4], q12 = Q[5];
            const float q20 = Q[6], q21 = Q[7], q22 = Q[8];

            // switch / chi
            const float inv_d = 1.0f / d;
            const float ddmp  = sqrtf(d * d + 1.0f);
            const float x     = d * 0.25f;          // d / CUTOFF_SR (exact)
            const float x2    = x * x;
            const float poly  = 10.0f - 15.0f * x + 6.0f * x2;
            const float s     = 1.0f - x2 * x * poly;
            const float sw    = (x < 1.0f) ? s : 0.0f;
            const float chi   = sw / ddmp + (1.0f - sw) * inv_d;

            const float chi2 = chi * chi;
            const float chi3 = chi2 * chi;
            const float chi_shift  = 0.2f   - 0.01f   * d;   // 2/C   - d/C^2
            const float chi2_shift = 0.03f  - 0.002f  * d;   // 3/C^2 - 2d/C^3
            const float chi3_shift = 0.004f - 0.0003f * d;   // 4/C^3 - 3d/C^4

            // charge-charge
            float E = qu * qv * (chi - chi_shift);

            // dipole terms
            const float nx = vx * inv_d, ny = vy * inv_d, nz = vz * inv_d;
            const float dot_uv = nx * dvx + ny * dvy + nz * dvz;
            const float dot_vu = nx * dux + ny * duy + nz * duz;
            const float dd_uv  = dux * dvx + duy * dvy + duz * dvz;
            E += 2.0f * qu * dot_uv * (chi2 - chi2_shift);
            E += (dd_uv - 3.0f * dot_uv * dot_vu) * (chi3 - chi3_shift);

            // quadrupole: sum(traceless(vv^T)/d^2 * Q) = (v^T Q v - (|v|^2/3) trQ)/d^2
            const float vQv =
                vx * (q00 * vx + q01 * vy + q02 * vz) +
                vy * (q10 * vx + q11 * vy + q12 * vz) +
                vz * (q20 * vx + q21 * vy + q22 * vz);
            const float trQ = q00 + q11 + q22;
            const float tm  = (vx * vx + vy * vy + vz * vz) * (1.0f / 3.0f);
            const float sum_uv = (vQv - tm * trQ) * inv_d * inv_d;
            E += qu * sum_uv * (chi3 - chi3_shift);

            E *= KEHALF;
            const float result = (d <= 10.0f) ? E : 0.0f;
            __builtin_nontemporal_store(result, out + e);
        }
        __syncthreads();                            // buf free for restage
        buf ^= 1;
    }
}

extern "C" void kernel_launch(void* const* d_in, const int* in_sizes, int n_in,
                              void* d_out, int out_size, void* d_ws, size_t ws_size,
                              hipStream_t stream) {
    const float* charges = (const float*)d_in[0];
    const float* dipoles = (const float*)d_in[1];
    const float* quads   = (const float*)d_in[2];
    const float* vec     = (const float*)d_in[3];
    const float* dist    = (const float*)d_in[4];
    const int*   idx_u   = (const int*)d_in[5];
    const int*   idx_v   = (const int*)d_in[6];
    float*       out     = (float*)d_out;

    const int n_edges = in_sizes[4];   // distances_uv element count == E

    const int block  = 256;            // 8 wave32 waves, one 256-edge tile
    int ntiles = (n_edges + block - 1) / block;
    int blocks = (ntiles > 2048) ? 2048 : ntiles;    // ~6 tiles/block pipeline
    if (blocks < 1) blocks = 1;

    hipLaunchKernelGGL(edge_elec_kernel, dim3(blocks), dim3(block), 0, stream,
                       charges, dipoles, quads, vec, dist, idx_u, idx_v, out,
                       n_edges);
}